// GAT_41472204210772
// MI455X (gfx1250) — compile-verified
//
#include <hip/hip_runtime.h>
#include <hip/hip_bf16.h>
#include <math.h>

typedef __attribute__((ext_vector_type(2))) float v2f;
typedef __attribute__((ext_vector_type(8))) float v8f;

#define NEG_SLOPE 0.2f

// ---- monotone float<->uint encoding for atomic float max ------------------
__device__ __forceinline__ unsigned fenc(float f) {
  unsigned u = __float_as_uint(f);
  return (u & 0x80000000u) ? ~u : (u | 0x80000000u);
}
__device__ __forceinline__ float fdec(unsigned u) {
  return __uint_as_float((u & 0x80000000u) ? (u & 0x7FFFFFFFu) : ~u);
}

// ---- init: zero accumulators (harness does not re-poison between replays) -
__global__ void gat_init(float* __restrict__ outb, float* __restrict__ ssum,
                         unsigned* __restrict__ mbuf, int N) {
  long long i = (long long)blockIdx.x * blockDim.x + threadIdx.x;
  long long total = (long long)N * 128;
  if (i < total) outb[i] = 0.0f;
  if (i < N) { ssum[i] = 0.0f; mbuf[i] = 0u; }  // enc(any finite) > 0
}

// ---- h = x @ W via V_WMMA_F32_16X16X4_F32 ---------------------------------
// Block: 256 thr = 8 waves. Block covers 16 rows of x; wave w covers cols
// [16w,16w+16). K loop steps 4 -> 64 WMMA issues per wave.
// A 16x4 f32 layout: lane l: M=l%16, holds K = (l/16)*2 .. +1 in 2 VGPRs.
// B 4x16 f32 layout (mirror of A over N): lane l: N=l%16, K=(l/16)*2 .. +1.
// C/D 16x16 f32: VGPR r: lanes 0-15 -> M=r, N=lane; lanes 16-31 -> M=r+8.
__global__ __launch_bounds__(256) void gat_gemm(const float* __restrict__ x,
                                                const float* __restrict__ W,
                                                float* __restrict__ h, int N) {
  const int lane = threadIdx.x & 31;
  const int wave = threadIdx.x >> 5;       // 0..7 -> N tile
  const int m0 = blockIdx.x << 4;          // 16 rows per block
  if (m0 >= N) return;
  const int r16 = lane & 15;
  const int khi = (lane >> 4) << 1;        // 0 or 2
  const int col = (wave << 4) + r16;       // 0..127
  const float* ap = x + (size_t)(m0 + r16) * 256 + khi;
  const float* bp = W + (size_t)khi * 128 + col;
  v8f acc = {};
  for (int kk = 0; kk < 256; kk += 4) {
    v2f a = *(const v2f*)ap;               // x[row][kk+khi], x[row][kk+khi+1]
    v2f b;
    b.x = bp[0];                           // W[kk+khi  ][col]
    b.y = bp[128];                         // W[kk+khi+1][col]
    acc = __builtin_amdgcn_wmma_f32_16x16x4_f32(false, a, false, b, (short)0,
                                                acc, false, false);
    ap += 4;
    bp += 4 * 128;
  }
  const int rbase = m0 + ((lane >> 4) << 3);
  float* hp = h + (size_t)rbase * 128 + col;
#pragma unroll
  for (int r = 0; r < 8; ++r) hp[(size_t)r * 128] = acc[r];
}

// ---- a_src[n] = h[n].att_src ; a_dst[n] = h[n].att_dst (wave per node) ----
__global__ __launch_bounds__(256) void gat_rowdots(const float* __restrict__ h,
                                                   const float* __restrict__ as,
                                                   const float* __restrict__ ad,
                                                   float* __restrict__ a_src,
                                                   float* __restrict__ a_dst,
                                                   int N) {
  const int node = blockIdx.x * 8 + (threadIdx.x >> 5);
  if (node >= N) return;
  const int lane = threadIdx.x & 31;
  const float* hr = h + (size_t)node * 128;
  float s0 = 0.0f, s1 = 0.0f;
#pragma unroll
  for (int j = 0; j < 4; ++j) {
    float v = hr[lane + 32 * j];
    s0 += v * as[lane + 32 * j];
    s1 += v * ad[lane + 32 * j];
  }
#pragma unroll
  for (int off = 16; off > 0; off >>= 1) {
    s0 += __shfl_xor(s0, off, 32);
    s1 += __shfl_xor(s1, off, 32);
  }
  if (lane == 0) { a_src[node] = s0; a_dst[node] = s1; }
}

// ---- edge pass 1: segment max of leaky_relu(a_src[s]+a_dst[d]) ------------
__global__ void gat_edge_max(const int* __restrict__ src, const int* __restrict__ dst,
                             const float* __restrict__ a_src,
                             const float* __restrict__ a_dst,
                             unsigned* __restrict__ mbuf, int E, int N) {
  const int i = blockIdx.x * blockDim.x + threadIdx.x;
  const int total = E + N;
  if (i >= total) return;
  int s, d;
  if (i < E) { s = src[i]; d = dst[i]; } else { s = d = i - E; }
  float e = a_src[s] + a_dst[d];
  e = (e > 0.0f) ? e : NEG_SLOPE * e;
  atomicMax(&mbuf[d], fenc(e));
}

// ---- edge pass 2: ex = exp(e - max[d]); segment sum -----------------------
__global__ void gat_edge_exp(const int* __restrict__ src, const int* __restrict__ dst,
                             const float* __restrict__ a_src,
                             const float* __restrict__ a_dst,
                             const unsigned* __restrict__ mbuf,
                             float* __restrict__ exbuf, float* __restrict__ ssum,
                             int E, int N) {
  const int i = blockIdx.x * blockDim.x + threadIdx.x;
  const int total = E + N;
  if (i >= total) return;
  int s, d;
  if (i < E) { s = src[i]; d = dst[i]; } else { s = d = i - E; }
  float e = a_src[s] + a_dst[d];
  e = (e > 0.0f) ? e : NEG_SLOPE * e;
  float ex = expf(e - fdec(mbuf[d]));
  exbuf[i] = ex;
  atomicAdd(&ssum[d], ex);
}

// ---- edge pass 3: out[d] += h[s] * alpha (wave per edge, float4 lanes) ----
__global__ __launch_bounds__(256) void gat_edge_scatter(
    const int* __restrict__ src, const int* __restrict__ dst,
    const float* __restrict__ h, const float* __restrict__ exbuf,
    const float* __restrict__ ssum, float* __restrict__ outb, int E, int N) {
  const int i = blockIdx.x * 8 + (threadIdx.x >> 5);
  const int total = E + N;
  if (i >= total) return;
  const int lane = threadIdx.x & 31;
  int s, d;
  if (i < E) { s = src[i]; d = dst[i]; } else { s = d = i - E; }
  const float alpha = exbuf[i] / (ssum[d] + 1e-16f);
  const float4 v = ((const float4*)(h + (size_t)s * 128))[lane];
  float* od = outb + (size_t)d * 128 + lane * 4;
  atomicAdd(&od[0], v.x * alpha);
  atomicAdd(&od[1], v.y * alpha);
  atomicAdd(&od[2], v.z * alpha);
  atomicAdd(&od[3], v.w * alpha);
}

// ---- final: (+bias_gat) -> FC(128->2) -> log_softmax (wave per node) ------
__global__ __launch_bounds__(256) void gat_final(const float* __restrict__ outb,
                                                 const float* __restrict__ bias_gat,
                                                 const float* __restrict__ fc_w,
                                                 const float* __restrict__ fc_b,
                                                 float* __restrict__ logout, int N) {
  const int node = blockIdx.x * 8 + (threadIdx.x >> 5);
  if (node >= N) return;
  const int lane = threadIdx.x & 31;
  const float* orow = outb + (size_t)node * 128;
  float d0 = 0.0f, d1 = 0.0f;
#pragma unroll
  for (int j = 0; j < 4; ++j) {
    int c = lane + 32 * j;
    float v = orow[c] + bias_gat[c];
    d0 += v * fc_w[c];         // fc_w row 0
    d1 += v * fc_w[128 + c];   // fc_w row 1
  }
#pragma unroll
  for (int off = 16; off > 0; off >>= 1) {
    d0 += __shfl_xor(d0, off, 32);
    d1 += __shfl_xor(d1, off, 32);
  }
  if (lane == 0) {
    float z0 = d0 + fc_b[0];
    float z1 = d1 + fc_b[1];
    float mx = fmaxf(z0, z1);
    float lse = mx + logf(expf(z0 - mx) + expf(z1 - mx));
    logout[(size_t)node * 2 + 0] = z0 - lse;
    logout[(size_t)node * 2 + 1] = z1 - lse;
  }
}

extern "C" void kernel_launch(void* const* d_in, const int* in_sizes, int n_in,
                              void* d_out, int out_size, void* d_ws, size_t ws_size,
                              hipStream_t stream) {
  const float* x        = (const float*)d_in[0];
  const int*   ei       = (const int*)d_in[1];
  const float* W        = (const float*)d_in[2];
  const float* att_src  = (const float*)d_in[3];
  const float* att_dst  = (const float*)d_in[4];
  const float* bias_gat = (const float*)d_in[5];
  const float* fc_w     = (const float*)d_in[6];
  const float* fc_b     = (const float*)d_in[7];
  float* logout = (float*)d_out;

  const int N = in_sizes[0] / 256;   // 100000
  const int E = in_sizes[1] / 2;     // 1600000
  const int* src = ei;
  const int* dst = ei + E;

  // workspace carve-up (256B aligned chunks)
  char* ws = (char*)d_ws;
  size_t off = 0;
  auto carve = [&](size_t bytes) -> void* {
    void* p = (void*)(ws + off);
    off += (bytes + 255) & ~(size_t)255;
    return p;
  };
  float*    h     = (float*)carve((size_t)N * 128 * sizeof(float));
  float*    outb  = (float*)carve((size_t)N * 128 * sizeof(float));
  float*    a_src = (float*)carve((size_t)N * sizeof(float));
  float*    a_dst = (float*)carve((size_t)N * sizeof(float));
  unsigned* mbuf  = (unsigned*)carve((size_t)N * sizeof(unsigned));
  float*    ssum  = (float*)carve((size_t)N * sizeof(float));
  float*    exbuf = (float*)carve((size_t)(E + N) * sizeof(float));

  const int total = E + N;

  {  // zero accumulators
    long long n = (long long)N * 128;
    int blocks = (int)((n + 255) / 256);
    gat_init<<<blocks, 256, 0, stream>>>(outb, ssum, mbuf, N);
  }
  {  // h = x @ W  (WMMA f32 16x16x4)
    int blocks = (N + 15) / 16;
    gat_gemm<<<blocks, 256, 0, stream>>>(x, W, h, N);
  }
  {  // a_src / a_dst
    int blocks = (N + 7) / 8;
    gat_rowdots<<<blocks, 256, 0, stream>>>(h, att_src, att_dst, a_src, a_dst, N);
  }
  {  // segment max
    int blocks = (total + 255) / 256;
    gat_edge_max<<<blocks, 256, 0, stream>>>(src, dst, a_src, a_dst, mbuf, E, N);
  }
  {  // exp + segment sum
    int blocks = (total + 255) / 256;
    gat_edge_exp<<<blocks, 256, 0, stream>>>(src, dst, a_src, a_dst, mbuf, exbuf,
                                             ssum, E, N);
  }
  {  // weighted scatter
    int blocks = (total + 7) / 8;
    gat_edge_scatter<<<blocks, 256, 0, stream>>>(src, dst, h, exbuf, ssum, outb,
                                                 E, N);
  }
  {  // bias + FC + log_softmax
    int blocks = (N + 7) / 8;
    gat_final<<<blocks, 256, 0, stream>>>(outb, bias_gat, fc_w, fc_b, logout, N);
  }
}